// Gate_29867202576995
// MI455X (gfx1250) — compile-verified
//
#include <hip/hip_runtime.h>
#include <hip/hip_bf16.h>

typedef __attribute__((ext_vector_type(2))) float v2f;
typedef __attribute__((ext_vector_type(8))) float v8f;
typedef __attribute__((ext_vector_type(4))) unsigned int u32x4;
typedef __attribute__((ext_vector_type(8))) int i32x8;
typedef __attribute__((ext_vector_type(4))) int i32x4;

#define D_DIM   2048
#define NE      64
#define TOPK    8
#define MROWS   16384          // B*T = 4*4096
#define MBLK    32             // rows per workgroup
#define KC      64             // K chunk
#define LDA     68             // padded LDS stride (dwords): 64 data + 4 pad (TDM pad codes 5/3)
#define LDS_S   65             // padded stride for scores tile
#define NCHUNK  (D_DIM / KC)   // 32

// Issue one TDM 2D tile load: tile_d1 rows x tile_d0 elems (f32) from a
// row-major tensor [tensor_d1 x tensor_d0] into LDS at lds_off, inserting
// 4 dwords of padding after every 64 dwords stored (-> LDA=68 row stride).
__device__ __forceinline__ void tdm_load_2d(unsigned int lds_off,
                                            const float* gptr,
                                            unsigned int tensor_d0,
                                            unsigned int tensor_d1,
                                            unsigned int tile_d0,
                                            unsigned int tile_d1,
                                            unsigned int stride0)
{
    unsigned long long ga = (unsigned long long)(uintptr_t)gptr;
    u32x4 g0;
    g0[0] = 1u;                                   // count=1 (valid), user mode
    g0[1] = lds_off;                              // LDS byte address
    g0[2] = (unsigned int)(ga & 0xFFFFFFFFu);     // global_addr[31:0]
    g0[3] = (unsigned int)((ga >> 32) & 0x01FFFFFFu) | (2u << 30); // addr[56:32] | type=2

    i32x8 g1;
    // data_size=4B (2), pad_enable, pad_interval=64dw (5), pad_amount=4dw (3)
    g1[0] = (int)((2u << 16) | (1u << 20) | (5u << 22) | (3u << 25));
    g1[1] = (int)((tensor_d0 & 0xFFFFu) << 16);                      // tensor_dim0[15:0]
    g1[2] = (int)((tensor_d0 >> 16) | ((tensor_d1 & 0xFFFFu) << 16)); // dim0[31:16] | dim1[15:0]
    g1[3] = (int)((tensor_d1 >> 16) | (tile_d0 << 16));               // dim1[31:16] | tile_dim0
    g1[4] = (int)tile_d1;                                             // tile_dim1 | tile_dim2=0
    g1[5] = (int)stride0;                                             // tensor_dim0_stride[31:0]
    g1[6] = 0;                                                        // stride[47:32] | dim1_stride lo
    g1[7] = 0;

    i32x4 z4 = {0, 0, 0, 0};                      // groups 2/3 unused (2D tensor)
    i32x8 z8 = {0, 0, 0, 0, 0, 0, 0, 0};
    __builtin_amdgcn_tensor_load_to_lds(g0, g1, z4, z4, z8, 0);
}

__global__ __launch_bounds__(256)
void moe_gate_kernel(const float* __restrict__ x,
                     const float* __restrict__ w,
                     const float* __restrict__ gate_b,
                     float* __restrict__ out_w,
                     int*   __restrict__ out_i,
                     float* __restrict__ out_s)
{
    __shared__ float lds_a[2][MBLK * LDA];   // double-buffered 32x64 X chunk
    __shared__ float lds_b[2][NE * LDA];     // double-buffered 64x64 W chunk
    __shared__ float lds_s[MBLK * LDS_S];    // 32x64 scores tile (padded)

    const int t      = threadIdx.x;
    const int wave   = t >> 5;
    const int lane   = t & 31;
    const int lq     = lane & 15;
    const int half   = lane >> 4;        // 0: K=k..k+1, 1: K=k+2..k+3
    const int m_sub  = wave >> 2;        // 0..1
    const int n_sub  = wave & 3;         // 0..3
    const int m0     = blockIdx.x * MBLK;

    v8f acc = {};

    // Prologue: wave 0 kicks the DMA for chunk 0.
    if (wave == 0) {
        tdm_load_2d((unsigned int)(uintptr_t)&lds_a[0][0],
                    x + (size_t)m0 * D_DIM, D_DIM, MROWS, KC, MBLK, D_DIM);
        tdm_load_2d((unsigned int)(uintptr_t)&lds_b[0][0],
                    w, D_DIM, NE, KC, NE, D_DIM);
    }

    for (int i = 0; i < NCHUNK; ++i) {
        const int buf = i & 1;
        // Publish chunk i: wave 0 drains its TENSORcnt, then barrier.
        if (wave == 0) {
            __builtin_amdgcn_s_wait_tensorcnt(0);
        }
        __syncthreads();

        // Prefetch chunk i+1 into the other buffer (its previous contents
        // were consumed in iteration i-1, which every wave finished before
        // passing the barrier above).
        if (wave == 0 && (i + 1) < NCHUNK) {
            const int k1 = (i + 1) * KC;
            tdm_load_2d((unsigned int)(uintptr_t)&lds_a[buf ^ 1][0],
                        x + (size_t)m0 * D_DIM + k1, D_DIM, MROWS, KC, MBLK, D_DIM);
            tdm_load_2d((unsigned int)(uintptr_t)&lds_b[buf ^ 1][0],
                        w + k1, D_DIM, NE, KC, NE, D_DIM);
        }

        // ---- 16 x WMMA f32 16x16x4 over this K chunk ----
        const float* pa = &lds_a[buf][(m_sub * 16 + lq) * LDA + half * 2];
        const float* pb = &lds_b[buf][(n_sub * 16 + lq) * LDA + half * 2];
        #pragma unroll
        for (int kk = 0; kk < KC; kk += 4) {
            v2f a = *(const v2f*)(pa + kk);
            v2f b = *(const v2f*)(pb + kk);
            acc = __builtin_amdgcn_wmma_f32_16x16x4_f32(
                false, a, false, b, (short)0, acc, false, false);
        }
        __syncthreads();   // all waves done with buf before TDM refills it next iter
    }

    // ---- epilogue: sigmoid + bias, scatter C tile to LDS scores ----
    // C layout (32-bit 16x16): lanes 0-15 -> M=v, lanes 16-31 -> M=8+v; N=lq
    {
        const int n = n_sub * 16 + lq;
        const float gb = gate_b[n];
        #pragma unroll
        for (int v = 0; v < 8; ++v) {
            const int m_local = m_sub * 16 + half * 8 + v;
            float s = 1.0f / (1.0f + __expf(-acc[v])) + gb;
            lds_s[m_local * LDS_S + n] = s;
        }
    }
    __syncthreads();

    // ---- write scores tile to global: out_s[(m0+r)*64 + c] ----
    {
        const int r = t >> 3;
        const int c = (t & 7) * 8;
        float* dst = out_s + (size_t)(m0 + r) * NE + c;
        #pragma unroll
        for (int i = 0; i < 8; ++i)
            dst[i] = lds_s[r * LDS_S + c + i];
    }

    // ---- top-8 per row (threads 0..31), serial first-max (matches lax.top_k ties) ----
    if (t < MBLK) {
        const int r = t;
        unsigned long long used = 0ull;
        float wv[TOPK];
        int   wi[TOPK];
        float wsum = 0.0f;
        #pragma unroll
        for (int sel = 0; sel < TOPK; ++sel) {
            float best = -3.4e38f;
            int   bi   = 0;
            for (int j = 0; j < NE; ++j) {
                if ((used >> j) & 1ull) continue;
                float v = lds_s[r * LDS_S + j];
                if (v > best) { best = v; bi = j; }
            }
            used |= (1ull << bi);
            wv[sel] = best;
            wi[sel] = bi;
            wsum += best;
        }
        const float inv = 1.0f / wsum;
        float* dw = out_w + (size_t)(m0 + r) * TOPK;
        int*   di = out_i + (size_t)(m0 + r) * TOPK;
        #pragma unroll
        for (int sel = 0; sel < TOPK; ++sel) {
            dw[sel] = wv[sel] * inv;
            di[sel] = wi[sel];
        }
    }
}

extern "C" void kernel_launch(void* const* d_in, const int* in_sizes, int n_in,
                              void* d_out, int out_size, void* d_ws, size_t ws_size,
                              hipStream_t stream) {
    const float* x      = (const float*)d_in[0];
    const float* w      = (const float*)d_in[1];
    const float* gate_b = (const float*)d_in[2];
    // d_in[3] = top_k scalar (8), compile-time constant here.

    float* out_w = (float*)d_out;                            // [16384, 8]
    int*   out_i = (int*)((float*)d_out + MROWS * TOPK);     // [16384, 8]
    float* out_s = (float*)d_out + (size_t)MROWS * TOPK * 2; // [16384, 64]

    dim3 grid(MROWS / MBLK);   // 512
    dim3 block(256);           // 8 waves
    moe_gate_kernel<<<grid, block, 0, stream>>>(x, w, gate_b, out_w, out_i, out_s);
}